// GlobalModel_11227044512396
// MI455X (gfx1250) — compile-verified
//
#include <hip/hip_runtime.h>

// ---------------------------------------------------------------------------
// MI455X (gfx1250) implementation.
//
// Phase 1: segment means over sorted batch indices.
//   - block b owns segment b (contiguous rows found via binary search)
//   - per-thread register accumulation (static unroll over F features)
//   - fixed-order LDS tree reduction -> deterministic, atomic-free
// Phase 2: tiny MLP [1024,25] @ W1 -> leaky_relu -> @ W2 using
//   V_WMMA_F32_16X16X4_F32 (full f32 precision matrix engine path).
// ---------------------------------------------------------------------------

typedef __attribute__((ext_vector_type(2))) float v2f;
typedef __attribute__((ext_vector_type(8))) float v8f;

// ---------------------------------------------------------------------------
// Kernel 1: per-segment mean. grid = nb blocks, 256 threads.
// x: [n, F] row-major f32; batch: [n] sorted int32 in [0, nb); mean: [nb, F].
// ---------------------------------------------------------------------------
template <int F>
__global__ __launch_bounds__(256) void seg_mean_kernel(
    const float* __restrict__ x, const int* __restrict__ batch, int n,
    float* __restrict__ mean) {
  const int b = blockIdx.x;

  // Uniform binary searches: lower_bound(b) and lower_bound(b+1).
  int lo = 0, hi = n;
  while (lo < hi) {
    int mid = (lo + hi) >> 1;
    if (batch[mid] < b) lo = mid + 1; else hi = mid;
  }
  int lo2 = lo, hi2 = n;
  while (lo2 < hi2) {
    int mid = (lo2 + hi2) >> 1;
    if (batch[mid] < b + 1) lo2 = mid + 1; else hi2 = mid;
  }
  const int start = lo, end = lo2;

  // Per-thread accumulation over a fixed strided partition of [start, end).
  float acc[F];
#pragma unroll
  for (int j = 0; j < F; ++j) acc[j] = 0.f;

  for (int r = start + (int)threadIdx.x; r < end; r += 256) {
#pragma unroll
    for (int j = 0; j < F; ++j) acc[j] += x[r * F + j];
  }

  __shared__ float red[256][F];
#pragma unroll
  for (int j = 0; j < F; ++j) red[threadIdx.x][j] = acc[j];
  __syncthreads();

  // Fixed-order tree reduction (deterministic).
  for (int off = 128; off > 0; off >>= 1) {
    if ((int)threadIdx.x < off) {
#pragma unroll
      for (int j = 0; j < F; ++j)
        red[threadIdx.x][j] += red[threadIdx.x + off][j];
    }
    __syncthreads();
  }

  if ((int)threadIdx.x < F) {
    float cnt = (float)(end - start);
    float d = cnt > 0.f ? cnt : 1.f;  // matches sums / max(cnt, 1)
    mean[b * F + threadIdx.x] = red[0][threadIdx.x] / d;
  }
}

// ---------------------------------------------------------------------------
// Kernel 2: MLP tail with f32 WMMA. One wave (32 threads) per 16-row tile.
//   out16x25 = [u | ms | mt] (pad K to 28)
//   h = leaky_relu(out @ W1 + b1, 0.1)   -> 7x wmma 16x16x4
//   y = h @ W2 + b2                      -> 4x wmma 16x16x4 (K=16)
// A/B lane layout: K = vgpr + 2*(lane>=16); C/D: M = vgpr + 8*(lane>=16),
// N = lane & 15 (per CDNA5 ISA 7.12.2).
// ---------------------------------------------------------------------------
__global__ __launch_bounds__(32) void mlp_wmma_kernel(
    const float* __restrict__ u, const float* __restrict__ W1,
    const float* __restrict__ b1, const float* __restrict__ W2,
    const float* __restrict__ b2, const float* __restrict__ ms,
    const float* __restrict__ mt, float* __restrict__ out) {
  __shared__ float inT[16][28];   // input tile, K padded 25->28
  __shared__ float W1p[28][16];   // W1 padded (zeros outside 25x10)
  __shared__ float W2p[16][16];   // W2 padded (zeros outside 10x10)
  __shared__ float hT[16][16];    // layer-1 activations
  __shared__ float b1s[16], b2s[16];

  const int tid = threadIdx.x;
  const int m0 = blockIdx.x * 16;

  for (int i = tid; i < 28 * 16; i += 32) {
    int r = i >> 4, c = i & 15;
    W1p[r][c] = (r < 25 && c < 10) ? W1[r * 10 + c] : 0.f;
  }
  for (int i = tid; i < 16 * 16; i += 32) {
    int r = i >> 4, c = i & 15;
    W2p[r][c] = (r < 10 && c < 10) ? W2[r * 10 + c] : 0.f;
  }
  if (tid < 16) {
    b1s[tid] = (tid < 10) ? b1[tid] : 0.f;
    b2s[tid] = (tid < 10) ? b2[tid] : 0.f;
  }
  // Input tile: cols 0..9 = u, 10..19 = ms, 20..24 = mt, 25..27 = 0.
  for (int i = tid; i < 16 * 28; i += 32) {
    int r = i / 28, k = i % 28;
    int row = m0 + r;
    float v;
    if (k < 10)      v = u[row * 10 + k];
    else if (k < 20) v = ms[row * 10 + (k - 10)];
    else if (k < 25) v = mt[row * 5 + (k - 20)];
    else             v = 0.f;
    inT[r][k] = v;
  }
  __syncthreads();

  const int half = tid >> 4;   // 0 or 1
  const int nm = tid & 15;     // M for A, N for B/C/D

  // Layer 1: acc(16x16) = inT(16x28) x W1p(28x16)
  v8f acc = {};
#pragma unroll
  for (int k0 = 0; k0 < 28; k0 += 4) {
    v2f a, bm;
    a.x = inT[nm][k0 + 2 * half + 0];
    a.y = inT[nm][k0 + 2 * half + 1];
    bm.x = W1p[k0 + 2 * half + 0][nm];
    bm.y = W1p[k0 + 2 * half + 1][nm];
    acc = __builtin_amdgcn_wmma_f32_16x16x4_f32(false, a, false, bm,
                                                (short)0, acc, false, false);
  }

  // bias + leaky_relu(0.1); stage h in LDS for layer-2 A operand.
#pragma unroll
  for (int v = 0; v < 8; ++v) {
    float hv = acc[v] + b1s[nm];
    hv = hv > 0.f ? hv : 0.1f * hv;
    hT[v + 8 * half][nm] = hv;
  }
  __syncthreads();

  // Layer 2: acc2(16x16) = hT(16x16) x W2p(16x16)
  v8f acc2 = {};
#pragma unroll
  for (int k0 = 0; k0 < 16; k0 += 4) {
    v2f a, bm;
    a.x = hT[nm][k0 + 2 * half + 0];
    a.y = hT[nm][k0 + 2 * half + 1];
    bm.x = W2p[k0 + 2 * half + 0][nm];
    bm.y = W2p[k0 + 2 * half + 1][nm];
    acc2 = __builtin_amdgcn_wmma_f32_16x16x4_f32(false, a, false, bm,
                                                 (short)0, acc2, false, false);
  }

  if (nm < 10) {
#pragma unroll
    for (int v = 0; v < 8; ++v) {
      int row = m0 + v + 8 * half;
      out[row * 10 + nm] = acc2[v] + b2s[nm];
    }
  }
}

// ---------------------------------------------------------------------------
// Launch.
// Inputs (setup_inputs order):
//   0 x_s[N_S,10] f32, 1 x_t[N_T,5] f32, 2 edge_attr (unused), 3 u[B,10] f32,
//   4 W1[25,10], 5 b1[10], 6 W2[10,10], 7 b2[10],
//   8 edge_index (unused), 9 batch_s[N_S] i32 sorted, 10 batch_t[N_T] i32.
// Output: [B,10] f32.
// ---------------------------------------------------------------------------
extern "C" void kernel_launch(void* const* d_in, const int* in_sizes, int n_in,
                              void* d_out, int out_size, void* d_ws,
                              size_t ws_size, hipStream_t stream) {
  const float* x_s = (const float*)d_in[0];
  const float* x_t = (const float*)d_in[1];
  const float* u   = (const float*)d_in[3];
  const float* W1  = (const float*)d_in[4];
  const float* b1  = (const float*)d_in[5];
  const float* W2  = (const float*)d_in[6];
  const float* b2  = (const float*)d_in[7];
  const int* batch_s = (const int*)d_in[9];
  const int* batch_t = (const int*)d_in[10];

  const int n_s = in_sizes[9];        // N_S
  const int n_t = in_sizes[10];       // N_T
  const int nb  = in_sizes[3] / 10;   // B (u is [B, 10])

  float* ms = (float*)d_ws;           // [nb, 10]
  float* mt = ms + (size_t)nb * 10;   // [nb, 5]

  seg_mean_kernel<10><<<nb, 256, 0, stream>>>(x_s, batch_s, n_s, ms);
  seg_mean_kernel<5><<<nb, 256, 0, stream>>>(x_t, batch_t, n_t, mt);

  mlp_wmma_kernel<<<nb / 16, 32, 0, stream>>>(u, W1, b1, W2, b2, ms, mt,
                                              (float*)d_out);
}